// MDA_83863531422299
// MI455X (gfx1250) — compile-verified
//
#include <hip/hip_runtime.h>
#include <hip/hip_bf16.h>

// ---------------------------------------------------------------------------
// Linear-MLP collapse for MI455X (gfx1250, wave32).
//   score(i,j) = s[i] + t[j] + b_eff
//   w_eff = W1 @ W2 @ W3 @ W4  (1802 floats),  s/t = sum_x @ w_eff halves
// Node-score matmul [1778 x 901] @ [901 x 2] runs on V_WMMA_F32_16X16X4_F32.
// ---------------------------------------------------------------------------

#define N_NODES 1778
#define DIM     901
#define H1      1024
#define H2      512
#define H3      64
#define NTRAIN  100000
#define NTEST   50000
#define NPAIRS  (NTRAIN + NTEST)

typedef __attribute__((ext_vector_type(2))) float v2f;
typedef __attribute__((ext_vector_type(8))) float v8f;

// ---- Step 1: v34[k] = sum_j W3[k,j] * W4[j]   (W3: [512,64] row-major) ----
__global__ __launch_bounds__(256) void k_v34(const float* __restrict__ W3,
                                             const float* __restrict__ W4,
                                             float* __restrict__ v34) {
    int k = blockIdx.x * blockDim.x + threadIdx.x;
    if (k >= H2) return;
    const float* row = W3 + (size_t)k * H3;
    float acc = 0.f;
    #pragma unroll 8
    for (int j = 0; j < H3; ++j) acc = fmaf(row[j], W4[j], acc);
    v34[k] = acc;
}

// ---- Step 2: v2[k] = sum_j W2[k,j] * v34[j]   (W2: [1024,512]) ------------
__global__ __launch_bounds__(256) void k_v2(const float* __restrict__ W2,
                                            const float* __restrict__ v34,
                                            float* __restrict__ v2) {
    int k = blockIdx.x * blockDim.x + threadIdx.x;
    if (k >= H1) return;
    const float* row = W2 + (size_t)k * H2;
    float acc = 0.f;
    #pragma unroll 8
    for (int j = 0; j < H2; ++j) acc = fmaf(row[j], v34[j], acc);
    v2[k] = acc;
}

// ---- Step 3: w_eff[k] = sum_j W1[k,j] * v2[j]  (W1: [1802,1024]) ----------
__global__ __launch_bounds__(256) void k_weff(const float* __restrict__ W1,
                                              const float* __restrict__ v2,
                                              float* __restrict__ weff) {
    int k = blockIdx.x * blockDim.x + threadIdx.x;
    if (k >= 2 * DIM) return;
    const float* row = W1 + (size_t)k * H1;
    float acc = 0.f;
    #pragma unroll 8
    for (int j = 0; j < H1; ++j) acc = fmaf(row[j], v2[j], acc);
    weff[k] = acc;
}

// ---- Step 4: b_eff = b4 + b3.W4 + b2.v34 + b1.v2 --------------------------
__global__ __launch_bounds__(256) void k_beff(const float* __restrict__ b1,
                                              const float* __restrict__ b2,
                                              const float* __restrict__ b3,
                                              const float* __restrict__ b4,
                                              const float* __restrict__ W4,
                                              const float* __restrict__ v34,
                                              const float* __restrict__ v2,
                                              float* __restrict__ beff) {
    __shared__ float red[256];
    int tid = threadIdx.x;
    float acc = 0.f;
    for (int j = tid; j < H1; j += 256) acc = fmaf(b1[j], v2[j], acc);
    for (int j = tid; j < H2; j += 256) acc = fmaf(b2[j], v34[j], acc);
    if (tid < H3) acc = fmaf(b3[tid], W4[tid], acc);
    red[tid] = acc;
    __syncthreads();
    for (int off = 128; off > 0; off >>= 1) {
        if (tid < off) red[tid] += red[tid + off];
        __syncthreads();
    }
    if (tid == 0) beff[0] = red[0] + b4[0];
}

// ---- Step 5: node scores via V_WMMA_F32_16X16X4_F32 -----------------------
// One wave32 per 16-row tile of sum_x. A = sum_x tile (16x4 f32 per step),
// B = w_eff columns padded 2 -> 16 (cols >= 2 are zero). All lane-dependent
// selection is branch-free so EXEC stays all-1s around the WMMAs.
__global__ __launch_bounds__(32) void k_node_scores(const float* __restrict__ sumx,
                                                    const float* __restrict__ weff,
                                                    float* __restrict__ s,
                                                    float* __restrict__ t) {
    const int m0   = blockIdx.x * 16;
    const int lane = threadIdx.x;      // 0..31
    const int half = lane >> 4;        // 0: K={0,1}, 1: K={2,3}
    const int l    = lane & 15;

    // A: row per lane (clamped; out-of-range rows are discarded at store time)
    int rowA = m0 + l;
    if (rowA >= N_NODES) rowA = N_NODES - 1;
    const float* arow = sumx + (size_t)rowA * DIM;

    // B: column per lane; only N=0,1 carry data
    const int   ncl   = (l < 2) ? l : 0;          // clamped column index
    const float nmask = (l < 2) ? 1.f : 0.f;
    const float* bcol = weff + (size_t)ncl * DIM;

    const int kb = half * 2;                       // 0 or 2
    v8f c = {};

    #pragma unroll 5
    for (int k = 0; k + 4 <= DIM; k += 4) {        // 225 full K-steps
        v2f a, b;
        a.x = arow[k + kb];
        a.y = arow[k + kb + 1];
        b.x = bcol[k + kb] * nmask;
        b.y = bcol[k + kb + 1] * nmask;
        c = __builtin_amdgcn_wmma_f32_16x16x4_f32(false, a, false, b,
                                                  (short)0, c, false, false);
    }
    // Remainder K-step: only k=900 is valid (half 0, first sub-element)
    {
        const float kmask = (half == 0) ? 1.f : 0.f;
        v2f a, b;
        a.x = arow[DIM - 1] * kmask;  a.y = 0.f;
        b.x = bcol[DIM - 1] * nmask * kmask;  b.y = 0.f;
        c = __builtin_amdgcn_wmma_f32_16x16x4_f32(false, a, false, b,
                                                  (short)0, c, false, false);
    }

    // D layout: VGPR r holds M = r (lanes 0-15) / r+8 (lanes 16-31), N = lane&15.
    if (l < 2) {
        float* dst = (l == 0) ? s : t;
        const int mbase = m0 + half * 8;
        #pragma unroll
        for (int r = 0; r < 8; ++r) {
            int m = mbase + r;
            if (m < N_NODES) dst[m] = c[r];
        }
    }
}

// ---- Step 6: out[p] = s[i] + t[j] + b_eff ---------------------------------
__global__ __launch_bounds__(256) void k_scores(const int* __restrict__ train,
                                                const int* __restrict__ test,
                                                const float* __restrict__ s,
                                                const float* __restrict__ t,
                                                const float* __restrict__ beff,
                                                float* __restrict__ out) {
    int p = blockIdx.x * blockDim.x + threadIdx.x;
    if (p >= NPAIRS) return;
    const int* pair = (p < NTRAIN) ? (train + 2 * (size_t)p)
                                   : (test + 2 * (size_t)(p - NTRAIN));
    out[p] = s[pair[0]] + t[pair[1]] + beff[0];
}

// ---------------------------------------------------------------------------
extern "C" void kernel_launch(void* const* d_in, const int* in_sizes, int n_in,
                              void* d_out, int out_size, void* d_ws, size_t ws_size,
                              hipStream_t stream) {
    const float* sum_x = (const float*)d_in[0];   // [1778, 901]
    const int*   train = (const int*)d_in[1];     // [100000, 2]
    const int*   test  = (const int*)d_in[2];     // [50000, 2]
    const float* W1    = (const float*)d_in[3];   // [1802, 1024]
    const float* b1    = (const float*)d_in[4];   // [1024]
    const float* W2    = (const float*)d_in[5];   // [1024, 512]
    const float* b2    = (const float*)d_in[6];   // [512]
    const float* W3    = (const float*)d_in[7];   // [512, 64]
    const float* b3    = (const float*)d_in[8];   // [64]
    const float* W4    = (const float*)d_in[9];   // [64, 1]
    const float* b4    = (const float*)d_in[10];  // [1]
    float* out = (float*)d_out;                   // 150000 floats

    // Workspace layout (floats)
    float* ws   = (float*)d_ws;
    float* v34  = ws;                 // 512
    float* v2   = v34 + 512;          // 1024
    float* weff = v2 + 1024;          // 1802
    float* beff = weff + 1802 + 6;    // 1 (padded to 16B boundary)
    float* sbuf = beff + 1 + 15;      // 1792 (1778 used)
    float* tbuf = sbuf + 1792;        // 1792 (1778 used)

    k_v34 <<<(H2 + 255) / 256, 256, 0, stream>>>(W3, W4, v34);
    k_v2  <<<(H1 + 255) / 256, 256, 0, stream>>>(W2, v34, v2);
    k_weff<<<(2 * DIM + 255) / 256, 256, 0, stream>>>(W1, v2, weff);
    k_beff<<<1, 256, 0, stream>>>(b1, b2, b3, b4, W4, v34, v2, beff);

    const int mtiles = (N_NODES + 15) / 16;       // 112 wave32 tiles
    k_node_scores<<<mtiles, 32, 0, stream>>>(sum_x, weff, sbuf, tbuf);

    k_scores<<<(NPAIRS + 255) / 256, 256, 0, stream>>>(train, test, sbuf, tbuf,
                                                       beff, out);
}